// single_channel_interp_91130616087185
// MI455X (gfx1250) — compile-verified
//
#include <hip/hip_runtime.h>
#include <hip/hip_bf16.h>
#include <cstdint>

typedef unsigned int u32x4 __attribute__((ext_vector_type(4)));
typedef unsigned int u32x8 __attribute__((ext_vector_type(8)));

#define Bq 8
#define Dq 64
#define Tq 512
#define Oq 256

__global__ __launch_bounds__(256)
void interp_softmax_kernel(const float* __restrict__ x,
                           const float* __restrict__ igrid,
                           const float* __restrict__ kern,
                           float* __restrict__ out)
{
    // LDS: [0..T) vals, [T..2T) mask -> A1 (in place), [2T..3T) times, [3T..4T) A10
    __shared__ float sh[4 * Tq];   // 8 KB of the 320 KB/WGP pool

    const int bd = blockIdx.x;
    const int b  = bd / Dq;
    const int d  = bd % Dq;
    const int tid = threadIdx.x;

    // ---------------- TDM: load 3 strided rows (vals, mask, times) into LDS ----------------
    // Rows live at channels d, D+d, 2D+d of x[b] -> base x + (b*3D + d)*T, row stride D*T elems.
    // 2-D tile: tile_dim0 = 512 elems (4B), tile_dim1 = 3 rows. Groups 2/3 omitted (2-D tensor).
    {
        unsigned lds_off = (unsigned)(unsigned long long)(const void*)&sh[0]; // flat addr[31:0] == LDS byte offset
        unsigned long long ga =
            (unsigned long long)(const void*)(x + ((size_t)(b * 3 * Dq + d) * Tq));

        u32x4 g0;
        g0[0] = 1u;                                            // count=1, user descriptor, gather off
        g0[1] = lds_off;                                       // lds_addr (bytes)
        g0[2] = (unsigned)(ga & 0xffffffffu);                  // global_addr[31:0]
        g0[3] = (unsigned)((ga >> 32) & 0x01ffffffu)           // global_addr[56:32]
              | (2u << 30);                                    // type = 2 ("image")

        u32x8 g1;
        g1[0] = (2u << 16);                                    // workgroup_mask=0, data_size=2 (4B), no flags
        g1[1] = ((unsigned)Tq & 0xffffu) << 16;                // atomic_barrier_addr=0 | tensor_dim0[15:0]
        g1[2] = (((unsigned)Tq >> 16) & 0xffffu)               // tensor_dim0[31:16]
              | (3u << 16);                                    // tensor_dim1[15:0] = 3
        g1[3] = ((unsigned)Tq & 0xffffu) << 16;                // tensor_dim1[31:16]=0 | tile_dim0 = 512
        g1[4] = 3u;                                            // tile_dim1 = 3, tile_dim2 = 0 (2-D)
        g1[5] = (unsigned)(Dq * Tq);                           // tensor_dim0_stride[31:0] = 32768 elems
        g1[6] = 0u;                                            // stride hi, tensor_dim1_stride lo (unused)
        g1[7] = 0u;

        asm volatile("tensor_load_to_lds %0, %1" :: "s"(g0), "s"(g1) : "memory");
        __builtin_amdgcn_s_wait_tensorcnt(0);
    }
    __syncthreads();

    float* __restrict__ sx   = sh;            // x_t values
    float* __restrict__ sA1  = sh + Tq;       // mask -> A1 = -a*t^2 + log(max(m,1e-30))
    float* __restrict__ st   = sh + 2 * Tq;   // times
    float* __restrict__ sA10 = sh + 3 * Tq;   // A10 = -10a*t^2 + log m

    // alpha = softplus(kernel[d]) (numerically stable)
    const float k = kern[d];
    const float alpha = fmaxf(k, 0.0f) + log1pf(expf(-fabsf(k)));

    // Precompute per-t terms cooperatively (each slot touched by exactly one thread).
    for (int i = tid; i < Tq; i += 256) {
        const float tv = st[i];
        const float lm = logf(fmaxf(sA1[i], 1e-30f));
        const float t2 = tv * tv;
        sA1[i]  = fmaf(-alpha,         t2, lm);
        sA10[i] = fmaf(-10.0f * alpha, t2, lm);
    }
    __syncthreads();

    // Each thread owns one grid point o. s[t] = A[t] + c1*t + c0 ; c0 folds out of softmax.
    const int   o   = tid;
    const float g   = igrid[b * Oq + o];
    const float c1  = 2.0f  * alpha * g;
    const float c1t = 20.0f * alpha * g;

    // Pass 1: maxes (LDS reads are same-address broadcasts across the wave -> no conflicts).
    float m1 = -3.4e38f, m10 = -3.4e38f;
#pragma unroll 8
    for (int t = 0; t < Tq; ++t) {
        const float tv = st[t];
        m1  = fmaxf(m1,  fmaf(c1,  tv, sA1[t]));
        m10 = fmaxf(m10, fmaf(c1t, tv, sA10[t]));
    }

    // Pass 2: exp2-based accumulation of Z and weighted sums for both temperatures.
    const float L2E = 1.4426950408889634f;
    const float b1  = -m1  * L2E;
    const float b10 = -m10 * L2E;
    float z1 = 0.0f, n1 = 0.0f, z10 = 0.0f, n10 = 0.0f;
#pragma unroll 4
    for (int t = 0; t < Tq; ++t) {
        const float tv = st[t];
        const float xv = sx[t];
        const float e1  = exp2f(fmaf(fmaf(c1,  tv, sA1[t]),  L2E, b1));
        const float e10 = exp2f(fmaf(fmaf(c1t, tv, sA10[t]), L2E, b10));
        z1  += e1;   n1  = fmaf(e1,  xv, n1);
        z10 += e10;  n10 = fmaf(e10, xv, n10);
    }

    // Outputs: concat([y, w, y_trans], axis=1) -> (B, 3D, O)
    const float  c0   = -alpha * g * g;
    const size_t base = ((size_t)b * (3 * Dq) + d) * Oq + o;
    out[base]                        = n1 / z1;                 // y
    out[base + (size_t)Dq * Oq]      = c0 + m1 + logf(z1);      // w = logsumexp
    out[base + (size_t)2 * Dq * Oq]  = n10 / z10;               // y_trans
}

extern "C" void kernel_launch(void* const* d_in, const int* in_sizes, int n_in,
                              void* d_out, int out_size, void* d_ws, size_t ws_size,
                              hipStream_t stream) {
    const float* x    = (const float*)d_in[0];   // (B, 3D, T) = 786432
    const float* grid = (const float*)d_in[1];   // (B, O)     = 2048
    const float* kern = (const float*)d_in[2];   // (D,)       = 64
    float* out        = (float*)d_out;           // (B, 3D, O) = 393216

    interp_softmax_kernel<<<dim3(Bq * Dq), dim3(Oq), 0, stream>>>(x, grid, kern, out);
}